// GAT_PYNQ_19808389169861
// MI455X (gfx1250) — compile-verified
//
#include <hip/hip_runtime.h>

// ---------------- problem constants (from reference) ----------------
#define N_NODES 50000
#define N_EDGES 800000
#define EP      (N_EDGES + N_NODES)   // edges + self loops = 850000
#define F_INF   128
#define HIDC    64
#define HEADS   4
#define NCLS    40
#define SELF_LOOP_W 4.0f
#define NEG_SLOPE   0.2f

typedef __attribute__((ext_vector_type(2))) float v2f;
typedef __attribute__((ext_vector_type(8))) float v8f;

__device__ __forceinline__ float leaky(float v) { return v > 0.f ? v : NEG_SLOPE * v; }

// float atomic max via int/uint trick (works for all signs; init to -inf)
__device__ __forceinline__ void atomicMaxF(float* addr, float val) {
  if (val >= 0.f) atomicMax((int*)addr, __float_as_int(val));
  else            atomicMin((unsigned int*)addr, __float_as_uint(val));
}

// ---------------- utility kernels ----------------
__global__ void k_fill(float* __restrict__ p, size_t n, float v) {
  size_t i = (size_t)blockIdx.x * blockDim.x + threadIdx.x;
  size_t st = (size_t)gridDim.x * blockDim.x;
  for (; i < n; i += st) p[i] = v;
}

__global__ void k_degree(const int* __restrict__ row, float* __restrict__ deg) {
  int e = blockIdx.x * blockDim.x + threadIdx.x;
  if (e < N_EDGES) atomicAdd(&deg[row[e]], 1.0f);
}

__global__ void k_dinv(float* __restrict__ deg) {
  int i = blockIdx.x * blockDim.x + threadIdx.x;
  if (i < N_NODES) deg[i] = rsqrtf(deg[i] + SELF_LOOP_W);  // always > 0
}

__global__ void k_norm(const int* __restrict__ row, const int* __restrict__ col,
                       const float* __restrict__ dinv, float* __restrict__ norm) {
  int e = blockIdx.x * blockDim.x + threadIdx.x;
  if (e >= EP) return;
  if (e < N_EDGES) norm[e] = dinv[row[e]] * dinv[col[e]];
  else { float d = dinv[e - N_EDGES]; norm[e] = SELF_LOOP_W * d * d; }
}

// ---------------- fp32 WMMA GEMM: C[M,Nn] = A[M,K] @ B[K,Nn] (+bias) ----------------
// one wave per 16(M) x 32(N) super-tile = two V_WMMA_F32_16X16X4_F32 tiles sharing the
// A fragment. M % 16 == 0, K % 4 == 0; Nn guarded branchlessly (clamped index + select,
// no EXEC divergence in the inner loop).
// Fragment layouts (ISA 7.12.2):
//   A 16x4 : lane = (k>=2)*16 + m ; vgpr = k&1   -> a = {A[m][2h], A[m][2h+1]}, h=lane>>4
//   B 4x16 : lane = (k>=2)*16 + n ; vgpr = k&1   -> b = {B[2h][n], B[2h+1][n]}
//   D 16x16: lane = n + 16*(m>=8) ; vgpr = m&7
__global__ void k_gemm_wmma_f32(const float* __restrict__ A, const float* __restrict__ B,
                                const float* __restrict__ bias, float* __restrict__ C,
                                int M, int Nn, int K) {
  int sTile = blockIdx.y * blockDim.y + threadIdx.y;   // 32-column super-tile
  if (sTile * 32 >= Nn) return;                        // uniform per wave
  int tileM = blockIdx.x;
  int lane = threadIdx.x;                              // 0..31
  int half = lane >> 4;
  int l15  = lane & 15;
  const float* Arow = A + (size_t)(tileM * 16 + l15) * K;
  int n0 = sTile * 32 + l15;
  int n1 = n0 + 16;
  bool ok0 = n0 < Nn, ok1 = n1 < Nn;
  int n0c = ok0 ? n0 : 0;                              // clamped (branchless guard)
  int n1c = ok1 ? n1 : 0;
  v8f acc0 = {0.f, 0.f, 0.f, 0.f, 0.f, 0.f, 0.f, 0.f};
  v8f acc1 = {0.f, 0.f, 0.f, 0.f, 0.f, 0.f, 0.f, 0.f};
#pragma unroll 4
  for (int k = 0; k < K; k += 4) {
    int ka = k + 2 * half;
    v2f a = *(const v2f*)(Arow + ka);                  // global_load_b64, 8B-aligned
    const float* Bk0 = B + (size_t)ka * Nn;
    const float* Bk1 = Bk0 + Nn;
    float x0 = Bk0[n0c], y0 = Bk1[n0c];
    float x1 = Bk0[n1c], y1 = Bk1[n1c];
    v2f b0, b1;
    b0.x = ok0 ? x0 : 0.f;  b0.y = ok0 ? y0 : 0.f;     // v_cndmask, no exec juggling
    b1.x = ok1 ? x1 : 0.f;  b1.y = ok1 ? y1 : 0.f;
    acc0 = __builtin_amdgcn_wmma_f32_16x16x4_f32(false, a, false, b0, (short)0, acc0,
                                                 false, false);
    acc1 = __builtin_amdgcn_wmma_f32_16x16x4_f32(false, a, false, b1, (short)0, acc1,
                                                 false, false);
  }
  int mbase = tileM * 16 + half * 8;
  if (ok0) {
    float bv = bias ? bias[n0] : 0.f;
#pragma unroll
    for (int r = 0; r < 8; ++r) C[(size_t)(mbase + r) * Nn + n0] = acc0[r] + bv;
  }
  if (ok1) {
    float bv = bias ? bias[n1] : 0.f;
#pragma unroll
    for (int r = 0; r < 8; ++r) C[(size_t)(mbase + r) * Nn + n1] = acc1[r] + bv;
  }
}

static void launch_gemm(const float* A, const float* B, const float* bias, float* C,
                        int M, int Nn, int K, hipStream_t s) {
  dim3 block(32, 4);
  int superTiles = (Nn + 31) / 32;
  dim3 grid(M / 16, (superTiles + 3) / 4);
  k_gemm_wmma_f32<<<grid, block, 0, s>>>(A, B, bias, C, M, Nn, K);
}

// ---------------- attention coefficient dot products ----------------
// es[n*H+h] = sum_c h[n,h,c]*a_src[h,c] ; ed likewise with a_dst
__global__ void k_attn(const float* __restrict__ h, const float* __restrict__ a_src,
                       const float* __restrict__ a_dst, float* __restrict__ es,
                       float* __restrict__ ed, int H, int C) {
  int idx = blockIdx.x * blockDim.x + threadIdx.x;
  if (idx >= N_NODES * H) return;
  int hh = idx % H, n = idx / H;
  const float* hp = h + (size_t)n * H * C + (size_t)hh * C;
  const float* as = a_src + hh * C;
  const float* ad = a_dst + hh * C;
  float s = 0.f, d = 0.f;
  for (int c = 0; c < C; ++c) { float v = hp[c]; s += v * as[c]; d += v * ad[c]; }
  es[idx] = s; ed[idx] = d;
}

// ---------------- edge pass 1: segment max of leaky-relu logits ----------------
__global__ void k_edge_max(const int* __restrict__ row, const int* __restrict__ col,
                           const float* __restrict__ es, const float* __restrict__ ed,
                           float* __restrict__ emax, int H) {
  int id = blockIdx.x * blockDim.x + threadIdx.x;
  if (id >= EP * H) return;
  int e = id / H, hh = id - e * H;
  int r = e < N_EDGES ? row[e] : e - N_EDGES;
  int c = e < N_EDGES ? col[e] : e - N_EDGES;
  float v = leaky(es[r * H + hh] + ed[c * H + hh]);
  atomicMaxF(&emax[r * H + hh], v);
}

// ---------------- edge pass 2: p=exp(e-max); esum += p; agg += p*norm*h[col] -------
// one wave (32 lanes) per edge; C must be 64 (channel>>6 head select).
__global__ void k_edge_msg(const int* __restrict__ row, const int* __restrict__ col,
                           const float* __restrict__ es, const float* __restrict__ ed,
                           const float* __restrict__ emax, float* __restrict__ esum,
                           const float* __restrict__ norm, const float* __restrict__ hsrc,
                           float* __restrict__ agg, int H) {
  int gid = blockIdx.x * blockDim.x + threadIdx.x;
  int wave = gid >> 5;
  int lane = gid & 31;
  if (wave >= EP) return;
  int r = wave < N_EDGES ? row[wave] : wave - N_EDGES;
  int c = wave < N_EDGES ? col[wave] : wave - N_EDGES;
  float nrm = norm[wave];
  float pl = 0.f;
  if (lane < H) {
    float v = leaky(es[r * H + lane] + ed[c * H + lane]);
    pl = expf(v - emax[r * H + lane]);
    atomicAdd(&esum[r * H + lane], pl);
  }
  float coef[HEADS];
#pragma unroll
  for (int hh = 0; hh < HEADS; ++hh) coef[hh] = __shfl(pl, hh, 32) * nrm;
  int HC = H * HIDC;
  const float* src = hsrc + (size_t)c * HC;
  float* dst = agg + (size_t)r * HC;
  for (int cc = lane; cc < HC; cc += 32)
    atomicAdd(&dst[cc], coef[cc >> 6] * src[cc]);   // C==64 -> head = cc/64
}

// ---------------- finalize: divide by softmax sum, optional relu ----------------
// total = N*H*C with C==64 ; esum indexed by (i>>6) = n*H+h
__global__ void k_finalize(float* __restrict__ agg, const float* __restrict__ esum,
                           size_t total, int do_relu) {
  size_t i = (size_t)blockIdx.x * blockDim.x + threadIdx.x;
  if (i >= total) return;
  float v = agg[i] / (esum[i >> 6] + 1e-16f);
  if (do_relu) v = fmaxf(v, 0.f);
  agg[i] = v;
}

// ---------------- host orchestration ----------------
extern "C" void kernel_launch(void* const* d_in, const int* in_sizes, int n_in,
                              void* d_out, int out_size, void* d_ws, size_t ws_size,
                              hipStream_t stream) {
  const float* x     = (const float*)d_in[0];   // [N, 128]
  const int*   eidx  = (const int*)d_in[1];     // [2, E]
  const float* W1    = (const float*)d_in[2];   // [128, 256]
  const float* asrc1 = (const float*)d_in[3];   // [4, 64]
  const float* adst1 = (const float*)d_in[4];   // [4, 64]
  const float* W2    = (const float*)d_in[5];   // [256, 64]
  const float* asrc2 = (const float*)d_in[6];   // [1, 64]
  const float* adst2 = (const float*)d_in[7];   // [1, 64]
  const float* linw  = (const float*)d_in[8];   // [64, 40]
  const float* linb  = (const float*)d_in[9];   // [40]
  const int* row = eidx;
  const int* col = eidx + N_EDGES;
  float* out = (float*)d_out;                   // [N, 40]

  // workspace layout (floats); layer-2 buffers alias layer-1 ones
  float* ws = (float*)d_ws;
  size_t o = 0;
  float* norm  = ws + o; o += (size_t)EP;
  float* dinv  = ws + o; o += (size_t)N_NODES;                 // deg then dinv in place
  float* h1    = ws + o; o += (size_t)N_NODES * HEADS * HIDC;  // also h2
  float* agg1  = ws + o; o += (size_t)N_NODES * HEADS * HIDC;  // also agg2
  float* es1   = ws + o; o += (size_t)N_NODES * HEADS;         // also es2
  float* ed1   = ws + o; o += (size_t)N_NODES * HEADS;         // also ed2
  float* emax1 = ws + o; o += (size_t)N_NODES * HEADS;         // also emax2
  float* esum1 = ws + o; o += (size_t)N_NODES * HEADS;         // also esum2
  float* h2 = h1; float* agg2 = agg1;
  float* es2 = es1; float* ed2 = ed1; float* emax2 = emax1; float* esum2 = esum1;

  const float NEG_INF = -__builtin_inff();
  const int B = 256;
  auto blocks = [](size_t n, int b) { return (unsigned)((n + b - 1) / b); };

  // -------- symmetric norm --------
  k_fill  <<<blocks(N_NODES, B), B, 0, stream>>>(dinv, N_NODES, 0.f);
  k_degree<<<blocks(N_EDGES, B), B, 0, stream>>>(row, dinv);
  k_dinv  <<<blocks(N_NODES, B), B, 0, stream>>>(dinv);
  k_norm  <<<blocks(EP, B), B, 0, stream>>>(row, col, dinv, norm);

  // -------- layer 1 (H=4, C=64) --------
  launch_gemm(x, W1, nullptr, h1, N_NODES, HEADS * HIDC, F_INF, stream);
  k_attn<<<blocks((size_t)N_NODES * HEADS, B), B, 0, stream>>>(h1, asrc1, adst1, es1, ed1,
                                                               HEADS, HIDC);
  k_fill<<<blocks((size_t)N_NODES * HEADS, B), B, 0, stream>>>(emax1, (size_t)N_NODES * HEADS, NEG_INF);
  k_fill<<<blocks((size_t)N_NODES * HEADS, B), B, 0, stream>>>(esum1, (size_t)N_NODES * HEADS, 0.f);
  k_fill<<<blocks((size_t)N_NODES * HEADS * HIDC, B), B, 0, stream>>>(agg1, (size_t)N_NODES * HEADS * HIDC, 0.f);
  k_edge_max<<<blocks((size_t)EP * HEADS, B), B, 0, stream>>>(row, col, es1, ed1, emax1, HEADS);
  k_edge_msg<<<blocks((size_t)EP * 32, B), B, 0, stream>>>(row, col, es1, ed1, emax1, esum1,
                                                           norm, h1, agg1, HEADS);
  k_finalize<<<blocks((size_t)N_NODES * HEADS * HIDC, B), B, 0, stream>>>(
      agg1, esum1, (size_t)N_NODES * HEADS * HIDC, /*relu=*/1);

  // -------- layer 2 (H=1, C=64) --------
  launch_gemm(agg1, W2, nullptr, h2, N_NODES, HIDC, HEADS * HIDC, stream);
  k_attn<<<blocks((size_t)N_NODES, B), B, 0, stream>>>(h2, asrc2, adst2, es2, ed2, 1, HIDC);
  k_fill<<<blocks((size_t)N_NODES, B), B, 0, stream>>>(emax2, (size_t)N_NODES, NEG_INF);
  k_fill<<<blocks((size_t)N_NODES, B), B, 0, stream>>>(esum2, (size_t)N_NODES, 0.f);
  k_fill<<<blocks((size_t)N_NODES * HIDC, B), B, 0, stream>>>(agg2, (size_t)N_NODES * HIDC, 0.f);
  k_edge_max<<<blocks((size_t)EP, B), B, 0, stream>>>(row, col, es2, ed2, emax2, 1);
  k_edge_msg<<<blocks((size_t)EP * 32, B), B, 0, stream>>>(row, col, es2, ed2, emax2, esum2,
                                                           norm, h2, agg2, 1);
  k_finalize<<<blocks((size_t)N_NODES * HIDC, B), B, 0, stream>>>(
      agg2, esum2, (size_t)N_NODES * HIDC, /*relu=*/0);

  // -------- classifier: out = agg2 @ lin_w + lin_b --------
  launch_gemm(agg2, linw, linb, out, N_NODES, NCLS, HIDC, stream);
}